// GraphConvLayer_47974784696352
// MI455X (gfx1250) — compile-verified
//
#include <hip/hip_runtime.h>
#include <math.h>

typedef float v2f __attribute__((ext_vector_type(2)));
typedef float v8f __attribute__((ext_vector_type(8)));

#define FEAT 128
#define BN_EPS 1e-3f

// ---------------- degree / normalization ----------------

__global__ void zero_f32(float* p, int n) {
    int i = blockIdx.x * blockDim.x + threadIdx.x;
    if (i < n) p[i] = 0.0f;
}

__global__ void deg_kernel(const int* __restrict__ rows, const float* __restrict__ w,
                           float* deg, int nE) {
    int e = blockIdx.x * blockDim.x + threadIdx.x;
    if (e < nE) atomicAdd(&deg[rows[e]], w[e]);
}

__global__ void dinv_kernel(const float* __restrict__ deg, float* __restrict__ dinv, int n) {
    int i = blockIdx.x * blockDim.x + threadIdx.x;
    if (i < n) {
        float d = deg[i] + 1.0f;            // + self-loop weight 1
        dinv[i] = (d > 0.0f) ? rsqrtf(d) : 0.0f;
    }
}

// ---------------- GEMM h = x @ W via V_WMMA_F32_16X16X4_F32 ----------------
// Block: 256 threads = 8 waves. Each wave computes a 16x128 tile of h.
// W (128x128 f32 = 64KB) staged in LDS once per block.

__global__ __launch_bounds__(256) void gemm_kernel(const float* __restrict__ x,
                                                   const float* __restrict__ W,
                                                   float* __restrict__ h, int nRows) {
    __shared__ float Wl[FEAT * FEAT];   // 64 KB

    int t = threadIdx.x;
    // Coalesced load of W into LDS (float4, 16 iters/thread)
    const float4* Wv = (const float4*)W;
    float4* Wlv = (float4*)Wl;
    #pragma unroll
    for (int i = t; i < (FEAT * FEAT) / 4; i += 256) Wlv[i] = Wv[i];
    __syncthreads();

    int wave = t >> 5;
    int lane = t & 31;
    int row0 = blockIdx.x * 128 + wave * 16;
    if (row0 >= nRows) return;          // wave-uniform exit (after the only barrier)

    // A-fragment addressing (ISA 7.12.2, 32-bit A 16x4):
    //   lanes 0-15  : row M=lane,    VGPR0=K0, VGPR1=K1
    //   lanes 16-31 : row M=lane-16, VGPR0=K2, VGPR1=K3
    int m  = lane & 15;
    int kh = (lane >> 4) << 1;          // 0 or 2
    int srow = row0 + m;
    if (srow >= nRows) srow = nRows - 1;    // clamp (stores masked later)
    const float* xr = x + (size_t)srow * FEAT;

    v8f acc[8] = {};                    // 8 column tiles of 16 -> 128 cols

    for (int kb = 0; kb < FEAT / 4; ++kb) {
        int k = kb * 4 + kh;
        float2 av = *(const float2*)(xr + k);
        v2f a; a.x = av.x; a.y = av.y;
        #pragma unroll
        for (int j = 0; j < 8; ++j) {
            int col = j * 16 + m;       // B 4x16: same K split across lane halves
            v2f b;
            b.x = Wl[(k + 0) * FEAT + col];
            b.y = Wl[(k + 1) * FEAT + col];
            acc[j] = __builtin_amdgcn_wmma_f32_16x16x4_f32(
                false, a, false, b, (short)0, acc[j], false, false);
        }
    }

    // C/D layout: VGPR r, lanes 0-15 -> M=r, lanes 16-31 -> M=r+8; N=lane%16
    int rbase = row0 + ((lane >> 4) ? 8 : 0);
    int n = lane & 15;
    #pragma unroll
    for (int j = 0; j < 8; ++j) {
        #pragma unroll
        for (int r = 0; r < 8; ++r) {
            int row = rbase + r;
            if (row < nRows) h[(size_t)row * FEAT + j * 16 + n] = acc[j][r];
        }
    }
}

// ---------------- self-loop init: out = h * dinv^2 ----------------

__global__ void self_loop_init(const float* __restrict__ h, const float* __restrict__ dinv,
                               float* __restrict__ out, int nNodes) {
    int i = blockIdx.x * blockDim.x + threadIdx.x;   // over nNodes * 32 float4 groups
    int total = nNodes * (FEAT / 4);
    if (i < total) {
        int node = i >> 5;                            // 32 float4 per node
        float s = dinv[node];
        s = s * s;
        float4 v = ((const float4*)h)[i];
        v.x *= s; v.y *= s; v.z *= s; v.w *= s;
        ((float4*)out)[i] = v;
    }
}

// ---------------- edge scatter: out[row] += h[col] * norm ----------------
// One wave per edge: lane covers 4 features (float4 gather, 4 f32 atomics).

__global__ __launch_bounds__(256) void edge_kernel(const int* __restrict__ rows,
                                                   const int* __restrict__ cols,
                                                   const float* __restrict__ w,
                                                   const float* __restrict__ dinv,
                                                   const float* __restrict__ h,
                                                   float* out, int nE) {
    int gid = blockIdx.x * blockDim.x + threadIdx.x;
    int wid = gid >> 5;
    int lane = gid & 31;
    int stride = (gridDim.x * blockDim.x) >> 5;
    for (int e = wid; e < nE; e += stride) {
        int r = rows[e];
        int c = cols[e];
        float norm = dinv[r] * w[e] * dinv[c];
        float4 hv = *(const float4*)(h + (size_t)c * FEAT + lane * 4);
        float* op = out + (size_t)r * FEAT + lane * 4;
        atomicAdd(op + 0, hv.x * norm);
        atomicAdd(op + 1, hv.y * norm);
        atomicAdd(op + 2, hv.z * norm);
        atomicAdd(op + 3, hv.w * norm);
    }
}

// ---------------- bias + BN(inference) + ReLU, in place ----------------

__global__ void bn_relu_kernel(float* __restrict__ out,
                               const float* __restrict__ bias,
                               const float* __restrict__ gamma,
                               const float* __restrict__ beta,
                               const float* __restrict__ mean,
                               const float* __restrict__ var, int total) {
    int i = blockIdx.x * blockDim.x + threadIdx.x;
    if (i < total) {
        int u = i & (FEAT - 1);
        float v = out[i] + bias[u];
        float r = gamma[u] * (v - mean[u]) * rsqrtf(var[u] + BN_EPS) + beta[u];
        out[i] = r > 0.0f ? r : 0.0f;
    }
}

// ---------------- launch ----------------

extern "C" void kernel_launch(void* const* d_in, const int* in_sizes, int n_in,
                              void* d_out, int out_size, void* d_ws, size_t ws_size,
                              hipStream_t stream) {
    const float* x     = (const float*)d_in[0];
    const int*   eidx  = (const int*)d_in[1];     // [2, E] row-major: rows then cols
    const float* ew    = (const float*)d_in[2];
    const float* W     = (const float*)d_in[3];
    const float* bias  = (const float*)d_in[4];
    const float* gamma = (const float*)d_in[5];
    const float* beta  = (const float*)d_in[6];
    const float* mean  = (const float*)d_in[7];
    const float* var   = (const float*)d_in[8];

    int nE = in_sizes[2];
    int nN = in_sizes[0] / FEAT;
    const int* rows = eidx;
    const int* cols = eidx + nE;

    float* h    = (float*)d_ws;                       // nN*FEAT floats
    float* deg  = h + (size_t)nN * FEAT;              // nN floats
    float* dinv = deg + nN;                           // nN floats
    float* out  = (float*)d_out;

    zero_f32<<<(nN + 255) / 256, 256, 0, stream>>>(deg, nN);
    deg_kernel<<<(nE + 255) / 256, 256, 0, stream>>>(rows, ew, deg, nE);
    dinv_kernel<<<(nN + 255) / 256, 256, 0, stream>>>(deg, dinv, nN);

    gemm_kernel<<<(nN + 127) / 128, 256, 0, stream>>>(x, W, h, nN);

    int initTotal = nN * (FEAT / 4);
    self_loop_init<<<(initTotal + 255) / 256, 256, 0, stream>>>(h, dinv, out, nN);

    long edgeThreads = (long)nE * 32;                 // one wave per edge
    edge_kernel<<<(int)((edgeThreads + 255) / 256), 256, 0, stream>>>(
        rows, cols, ew, dinv, h, out, nE);

    int total = nN * FEAT;
    bn_relu_kernel<<<(total + 255) / 256, 256, 0, stream>>>(
        out, bias, gamma, beta, mean, var, total);
}